// FCGAT_84842783965485
// MI455X (gfx1250) — compile-verified
//
#include <hip/hip_runtime.h>

typedef __attribute__((ext_vector_type(2))) float v2f;
typedef __attribute__((ext_vector_type(8))) float v8f;

#define NTOK 64    // K tokens per sample
#define LSX 132    // padded LDS stride for 128-wide matrices (bank-conflict free)
#define LSH 68     // padded LDS stride for 64-wide matrices

// LDS layout (floats)
#define OF_XC   0        // 64 x LSX   (towers | x)
#define OF_H1   8448     // 64 x LSX   (h1, later reused as LEFT)
#define OF_R    16896    // 64 x LSX   (RIGHT)
#define OF_H2   25344    // 64 x LSH   (h2)
#define OF_AA   29696    // 64 x 64    (attention scores)
#define OF_B1   33792    // 128
#define OF_B2   33920    // 64
#define OF_AB1  33984    // 128
#define OF_AW2  34112    // 128
#define OF_OW   34240    // 64
#define OF_RS   34304    // 64 (softmax row sums)
#define OF_RED  34368    // 64 (sigmoid product reduction)
#define SMEM_F  34432

__device__ __forceinline__ float lrelu(float v) {
    return fmaxf(v, 0.01f * v);   // exact for both signs (slope 0.01)
}

__global__ void __launch_bounds__(256)
fcgat_kernel(const float* __restrict__ towers, const float* __restrict__ x,
             const float* __restrict__ w1, const float* __restrict__ b1,
             const float* __restrict__ w2, const float* __restrict__ b2,
             const float* __restrict__ aw1, const float* __restrict__ ab1,
             const float* __restrict__ aw2, const float* __restrict__ ab2,
             const float* __restrict__ ow, const float* __restrict__ ob,
             const int* __restrict__ kiter,
             float* __restrict__ out)
{
    __shared__ float smem[SMEM_F];
    float* sXC  = smem + OF_XC;
    float* sH1  = smem + OF_H1;   // h1, then LEFT
    float* sR   = smem + OF_R;
    float* sH2  = smem + OF_H2;
    float* sAA  = smem + OF_AA;
    float* sB1  = smem + OF_B1;
    float* sB2  = smem + OF_B2;
    float* sAB1 = smem + OF_AB1;
    float* sAW2 = smem + OF_AW2;
    float* sOW  = smem + OF_OW;
    float* sRS  = smem + OF_RS;
    float* sRED = smem + OF_RED;

    const int tid  = threadIdx.x;
    const int n    = blockIdx.x;
    const int wv   = tid >> 5;         // wave id 0..7 (wave32)
    const int lane = tid & 31;
    const int lo   = lane & 15;        // fragment row/col within 16
    const int hi   = lane >> 4;        // fragment K-half select

    // ---- stage small vectors ----
    if (tid < 128) { sB1[tid] = b1[tid]; sAB1[tid] = ab1[tid]; sAW2[tid] = aw2[tid]; }
    if (tid < 64)  { sB2[tid] = b2[tid]; sOW[tid] = ow[tid]; }

    // ---- stage xc = [towers | x] (64 x 128) ----
    {
        const float4* t4 = (const float4*)(towers + (size_t)n * NTOK * 64);
        const float4* x4 = (const float4*)(x      + (size_t)n * NTOK * 64);
        for (int idx = tid; idx < NTOK * 16; idx += 256) {
            int m = idx >> 4, c = idx & 15;
            *(float4*)&sXC[m * LSX + c * 4]      = t4[m * 16 + c];
            *(float4*)&sXC[m * LSX + 64 + c * 4] = x4[m * 16 + c];
        }
    }
    __syncthreads();

    const int K = kiter[0];
    for (int it = 0; it < K; ++it) {
        // ======== GEMM1: h1 = lrelu(XC(64x128) @ w1^T + b1)  -> sH1 ========
        {
            const int m0 = (wv >> 1) * 16;
            const int n0 = (wv & 1) * 64;
            v8f acc[4] = {};
            for (int k = 0; k < 128; k += 4) {
                v2f a = *(const v2f*)&sXC[(m0 + lo) * LSX + k + 2 * hi];
#pragma unroll
                for (int j = 0; j < 4; ++j) {
                    v2f b = *(const v2f*)&w1[(n0 + j * 16 + lo) * 128 + k + 2 * hi];
                    acc[j] = __builtin_amdgcn_wmma_f32_16x16x4_f32(
                        false, a, false, b, (short)0, acc[j], false, false);
                }
            }
#pragma unroll
            for (int j = 0; j < 4; ++j) {
                int col = n0 + j * 16 + lo;
                float bb = sB1[col];
#pragma unroll
                for (int r = 0; r < 8; ++r)
                    sH1[(m0 + r + 8 * hi) * LSX + col] = lrelu(acc[j][r] + bb);
            }
        }
        __syncthreads();

        // ======== GEMM2: h2 = lrelu(H1(64x128) @ w2^T + b2) -> sH2 ========
        {
            const int m0 = (wv >> 1) * 16;
            const int n0 = (wv & 1) * 32;
            v8f acc[2] = {};
            for (int k = 0; k < 128; k += 4) {
                v2f a = *(const v2f*)&sH1[(m0 + lo) * LSX + k + 2 * hi];
#pragma unroll
                for (int j = 0; j < 2; ++j) {
                    v2f b = *(const v2f*)&w2[(n0 + j * 16 + lo) * 128 + k + 2 * hi];
                    acc[j] = __builtin_amdgcn_wmma_f32_16x16x4_f32(
                        false, a, false, b, (short)0, acc[j], false, false);
                }
            }
            __syncthreads();   // all reads of sH1 done before it becomes LEFT
#pragma unroll
            for (int j = 0; j < 2; ++j) {
                int col = n0 + j * 16 + lo;
                float bb = sB2[col];
#pragma unroll
                for (int r = 0; r < 8; ++r)
                    sH2[(m0 + r + 8 * hi) * LSH + col] = lrelu(acc[j][r] + bb);
            }
        }
        __syncthreads();

        // ======== GEMM3+4: LEFT = H2 @ aw1[:,:64]^T ; RIGHT = H2 @ aw1[:,64:]^T ====
        {
            const int m0 = (wv >> 1) * 16;
            const int n0 = (wv & 1) * 64;
            v8f accL[4] = {};
            v8f accR[4] = {};
            for (int k = 0; k < 64; k += 4) {
                v2f a = *(const v2f*)&sH2[(m0 + lo) * LSH + k + 2 * hi];
#pragma unroll
                for (int j = 0; j < 4; ++j) {
                    v2f bl = *(const v2f*)&aw1[(n0 + j * 16 + lo) * 128 + k + 2 * hi];
                    accL[j] = __builtin_amdgcn_wmma_f32_16x16x4_f32(
                        false, a, false, bl, (short)0, accL[j], false, false);
                    v2f br = *(const v2f*)&aw1[(n0 + j * 16 + lo) * 128 + 64 + k + 2 * hi];
                    accR[j] = __builtin_amdgcn_wmma_f32_16x16x4_f32(
                        false, a, false, br, (short)0, accR[j], false, false);
                }
            }
#pragma unroll
            for (int j = 0; j < 4; ++j) {
                int col = n0 + j * 16 + lo;
#pragma unroll
                for (int r = 0; r < 8; ++r) {
                    sH1[(m0 + r + 8 * hi) * LSX + col] = accL[j][r];  // LEFT
                    sR [(m0 + r + 8 * hi) * LSX + col] = accR[j][r];  // RIGHT
                }
            }
        }
        __syncthreads();

        // ======== pairwise scores: aa[i,j] = lrelu(sum_h lrelu(L+R+ab1)*aw2 + ab2) ==
        {
            const float ab2s = ab2[0];
            for (int q = 0; q < 16; ++q) {
                int p = q * 256 + tid;
                int i = p >> 6, j = p & 63;
                const float* Lr = &sH1[i * LSX];
                const float* Rr = &sR[j * LSX];
                float acc = 0.f;
#pragma unroll 8
                for (int h = 0; h < 128; ++h) {
                    float z = Lr[h] + Rr[h] + sAB1[h];
                    acc = fmaf(lrelu(z), sAW2[h], acc);
                }
                sAA[p] = lrelu(acc + ab2s);
            }
        }
        __syncthreads();

        // ======== softmax row-sums (einsum contracts only j => out = h2*rowsum) ====
        if (tid < 64) {
            const float* row = &sAA[tid * 64];
            float m = row[0];
            for (int j = 1; j < 64; ++j) m = fmaxf(m, row[j]);
            float s = 0.f;
            for (int j = 0; j < 64; ++j) s += expf(row[j] - m);
            float inv = 1.0f / s;
            float rs = 0.f;
            for (int j = 0; j < 64; ++j) rs += expf(row[j] - m) * inv;
            sRS[tid] = rs;
        }
        __syncthreads();

        // ======== residual update: x <- h2*rowsum + x_old (in place in sXC) ========
        for (int idx = tid; idx < 64 * 64; idx += 256) {
            int i = idx >> 6, d = idx & 63;
            float xo = sXC[i * LSX + 64 + d];
            sXC[i * LSX + 64 + d] = fmaf(sH2[i * LSH + d], sRS[i], xo);
        }
        __syncthreads();
    }

    // ======== logits -> sigmoid -> product over tokens ========
    if (tid < 64) {
        float acc = ob[0];
        for (int d = 0; d < 64; ++d) acc = fmaf(sXC[tid * LSX + 64 + d], sOW[d], acc);
        sRED[tid] = 1.0f / (1.0f + expf(-acc));
    }
    __syncthreads();
    if (tid == 0) {
        float p = 1.0f;
        for (int i = 0; i < 64; ++i) p *= sRED[i];
        out[n] = p;
    }
}

extern "C" void kernel_launch(void* const* d_in, const int* in_sizes, int n_in,
                              void* d_out, int out_size, void* d_ws, size_t ws_size,
                              hipStream_t stream) {
    const float* towers = (const float*)d_in[0];
    const float* x      = (const float*)d_in[1];
    const float* w1     = (const float*)d_in[2];
    const float* b1     = (const float*)d_in[3];
    const float* w2     = (const float*)d_in[4];
    const float* b2     = (const float*)d_in[5];
    const float* aw1    = (const float*)d_in[6];
    const float* ab1    = (const float*)d_in[7];
    const float* aw2    = (const float*)d_in[8];
    const float* ab2    = (const float*)d_in[9];
    const float* ow     = (const float*)d_in[10];
    const float* ob     = (const float*)d_in[11];
    const int*   kit    = (const int*)d_in[12];
    float* outp = (float*)d_out;

    const int N = out_size;  // one sample per workgroup
    fcgat_kernel<<<N, 256, 0, stream>>>(towers, x, w1, b1, w2, b2,
                                        aw1, ab1, aw2, ab2, ow, ob, kit, outp);
}